// ElasticConv_51505247814277
// MI455X (gfx1250) — compile-verified
//
#include <hip/hip_runtime.h>

#define NN   100000          // nodes
#define NE   800000          // undirected edges
#define MD   1600000         // directed edges
#define NF   (NN * 64)       // node-feature floats
#define GAMMA        0.25f   // 1/(1+lambda2)
#define ONE_M_GAMMA  0.75f
#define BETA         2.0f    // 1/(2*gamma)
#define LAM1         3.0f

typedef __attribute__((ext_vector_type(2))) float v2f;
typedef __attribute__((ext_vector_type(8))) float v8f;

// ---------------- degree / dinv ----------------
__global__ void k_deg_init(float* __restrict__ deg) {
    int i = blockIdx.x * blockDim.x + threadIdx.x;
    if (i < NN) deg[i] = 1.0f;               // self loop
}

__global__ void k_deg_count(const int* __restrict__ ei, float* __restrict__ deg) {
    int e = blockIdx.x * blockDim.x + threadIdx.x;
    if (e < MD) atomicAdd(&deg[ei[e]], 1.0f);
}

__global__ void k_dinv(float* __restrict__ deg) {
    int i = blockIdx.x * blockDim.x + threadIdx.x;
    if (i < NN) deg[i] = 1.0f / sqrtf(deg[i]);   // in place: deg -> dinv
}

// ---------------- y = gamma*hh + (1-gamma)*dinv^2*x  (self-loop part) ----------------
__global__ void k_yinit(const float* __restrict__ hh, const float* __restrict__ x,
                        const float* __restrict__ dinv, float* __restrict__ y) {
    int t = blockIdx.x * blockDim.x + threadIdx.x;     // one float4 per thread
    if (t >= NF / 4) return;
    int node = t >> 4;
    float w = dinv[node];
    w = ONE_M_GAMMA * w * w;
    const float4 h  = ((const float4*)hh)[t];
    const float4 xv = ((const float4*)x)[t];
    float4 r;
    r.x = GAMMA * h.x + w * xv.x;
    r.y = GAMMA * h.y + w * xv.y;
    r.z = GAMMA * h.z + w * xv.z;
    r.w = GAMMA * h.w + w * xv.w;
    ((float4*)y)[t] = r;
}

// ---------------- y[r] += (1-gamma)*dinv[r]*dinv[c]*x[c]  over directed edges ----------------
__global__ void k_spmm(const int* __restrict__ ei, const float* __restrict__ x,
                       const float* __restrict__ dinv, float* __restrict__ y) {
    int t = blockIdx.x * blockDim.x + threadIdx.x;     // 16 threads per edge, float4 each
    if (t >= MD * 16) return;
    int e = t >> 4;
    int q = (t & 15) << 2;
    int r = ei[e];
    int c = ei[MD + e];
    float w = ONE_M_GAMMA * dinv[r] * dinv[c];
    const float4 xv = *(const float4*)(x + (size_t)c * 64 + q);
    float* yp = y + (size_t)r * 64 + q;
    atomicAdd(yp + 0, w * xv.x);
    atomicAdd(yp + 1, w * xv.y);
    atomicAdd(yp + 2, w * xv.z);
    atomicAdd(yp + 3, w * xv.w);
}

// ---------------- xb = y ; xn = y ----------------
__global__ void k_copy2(const float* __restrict__ y, float* __restrict__ xb,
                        float* __restrict__ xn) {
    int t = blockIdx.x * blockDim.x + threadIdx.x;
    if (t >= NF / 4) return;
    float4 v = ((const float4*)y)[t];
    ((float4*)xb)[t] = v;
    ((float4*)xn)[t] = v;
}

// ---------------- xb[hi] -= g*w_hi*z ; xb[lo] += g*w_lo*z  (x_bar scatter) ----------------
__global__ void k_ztscatter(const int* __restrict__ ei, const float* __restrict__ z,
                            const float* __restrict__ dinv, float* __restrict__ xb) {
    int t = blockIdx.x * blockDim.x + threadIdx.x;     // 16 threads per edge
    if (t >= NE * 16) return;
    int e = t >> 4;
    int q = (t & 15) << 2;
    int u = ei[e];
    int v = ei[MD + e];
    int hi = u > v ? u : v;
    int lo = u > v ? v : u;
    float wh = GAMMA * dinv[hi];
    float wl = GAMMA * dinv[lo];
    const float4 zv = *(const float4*)(z + (size_t)e * 64 + q);
    float* ph = xb + (size_t)hi * 64 + q;
    float* pl = xb + (size_t)lo * 64 + q;
    atomicAdd(ph + 0, -wh * zv.x);
    atomicAdd(ph + 1, -wh * zv.y);
    atomicAdd(ph + 2, -wh * zv.z);
    atomicAdd(ph + 3, -wh * zv.w);
    atomicAdd(pl + 0,  wl * zv.x);
    atomicAdd(pl + 1,  wl * zv.y);
    atomicAdd(pl + 2,  wl * zv.z);
    atomicAdd(pl + 3,  wl * zv.w);
}

// ---------------- fused: z_bar -> L2,1 projection (WMMA row norms) -> z_new, xn scatter ----
// One wave processes 16 edges. A-matrix layout (f32 16x4): lane<16 holds K=0,1 (v0,v1),
// lane>=16 holds K=2,3. So lane L covers edge (L&15), features {4c+2*(L>>4), +1}.
// B = all ones -> D[m,*] = sum_k A[m,k]; accumulate 16 chunks = squared row norm.
template <bool ZZERO>
__global__ void k_zupdate(const int* __restrict__ ei, const float* __restrict__ xb,
                          const float* __restrict__ dinv, float* __restrict__ z,
                          float* __restrict__ xn) {
    int wave = (blockIdx.x * blockDim.x + threadIdx.x) >> 5;
    int lane = threadIdx.x & 31;
    int te = wave * 16;
    if (te >= NE) return;                  // never taken (NE % 16 == 0), wave-uniform
    int el   = lane & 15;
    int half = lane >> 4;
    int e = te + el;
    int u = ei[e];
    int v = ei[MD + e];
    int hi = u > v ? u : v;
    int lo = u > v ? v : u;
    float wh = dinv[hi];
    float wl = dinv[lo];
    const float* xh = xb + (size_t)hi * 64;
    const float* xl = xb + (size_t)lo * 64;
    __builtin_prefetch(xh, 0, 0);
    __builtin_prefetch(xl, 0, 0);
    float* zp = z + (size_t)e * 64;
    int off0 = half * 2;

    float2 zb[16];
    v8f acc = {0.f, 0.f, 0.f, 0.f, 0.f, 0.f, 0.f, 0.f};
    v2f ones;
    ones.x = 1.0f; ones.y = 1.0f;

#pragma unroll
    for (int c = 0; c < 16; ++c) {
        int off = c * 4 + off0;
        float z0 = ZZERO ? 0.0f : zp[off];
        float z1 = ZZERO ? 0.0f : zp[off + 1];
        float zb0 = z0 + BETA * (wh * xh[off]     - wl * xl[off]);
        float zb1 = z1 + BETA * (wh * xh[off + 1] - wl * xl[off + 1]);
        zb[c].x = zb0;
        zb[c].y = zb1;
        v2f a;
        a.x = zb0 * zb0;
        a.y = zb1 * zb1;
        acc = __builtin_amdgcn_wmma_f32_16x16x4_f32(
            false, a, false, ones, (short)0, acc, false, false);
    }

    // D layout: lanes 0-15 hold rows 0..7 in acc[0..7]; lanes 16-31 hold rows 8..15.
    int e3 = el & 7;
    float a0 = (e3 & 1) ? acc[1] : acc[0];
    float a1 = (e3 & 1) ? acc[3] : acc[2];
    float a2 = (e3 & 1) ? acc[5] : acc[4];
    float a3 = (e3 & 1) ? acc[7] : acc[6];
    float b0 = (e3 & 2) ? a1 : a0;
    float b1 = (e3 & 2) ? a3 : a2;
    float sel = (e3 & 4) ? b1 : b0;
    int src = (el < 8) ? el : (el + 8);
    float rn2 = __shfl(sel, src, 32);

    float scale = 0.0f;
    if (rn2 > 0.0f) {
        float rn = sqrtf(rn2);
        scale = fminf(rn, LAM1) / rn;
    }

    float gwh = GAMMA * wh;
    float gwl = GAMMA * wl;
    float* xnh = xn + (size_t)hi * 64;
    float* xnl = xn + (size_t)lo * 64;
#pragma unroll
    for (int c = 0; c < 16; ++c) {
        int off = c * 4 + off0;
        float zn0 = scale * zb[c].x;
        float zn1 = scale * zb[c].y;
        zp[off]     = zn0;
        zp[off + 1] = zn1;
        atomicAdd(xnh + off,     -gwh * zn0);
        atomicAdd(xnh + off + 1, -gwh * zn1);
        atomicAdd(xnl + off,      gwl * zn0);
        atomicAdd(xnl + off + 1,  gwl * zn1);
    }
}

extern "C" void kernel_launch(void* const* d_in, const int* in_sizes, int n_in,
                              void* d_out, int out_size, void* d_ws, size_t ws_size,
                              hipStream_t stream) {
    const float* x0 = (const float*)d_in[0];       // (N, 64)
    const int*   ei = (const int*)d_in[1];         // (2, 1.6M) row-major
    float* out = (float*)d_out;

    float* f    = (float*)d_ws;
    float* dinv = f;                               // NN floats (padded to 102400)
    float* y    = f + 102400;                      // NF
    float* xb   = y + NF;                          // NF
    float* xA   = xb + NF;                         // NF
    float* z    = xA + NF;                         // NE*64

    const int B = 256;
    k_deg_init<<<(NN + B - 1) / B, B, 0, stream>>>(dinv);
    k_deg_count<<<MD / B, B, 0, stream>>>(ei, dinv);
    k_dinv<<<(NN + B - 1) / B, B, 0, stream>>>(dinv);

    const float* xin = x0;
    for (int s = 0; s < 3; ++s) {
        float* xn = (s == 2) ? out : xA;
        k_yinit<<<(NF / 4) / B, B, 0, stream>>>(x0, xin, dinv, y);
        k_spmm<<<(MD * 16) / B, B, 0, stream>>>(ei, xin, dinv, y);
        k_copy2<<<(NF / 4) / B, B, 0, stream>>>(y, xb, xn);
        if (s > 0)
            k_ztscatter<<<(NE * 16) / B, B, 0, stream>>>(ei, z, dinv, xb);
        if (s == 0)
            k_zupdate<true><<<(NE / 16 * 32) / B, B, 0, stream>>>(ei, xb, dinv, z, xn);
        else
            k_zupdate<false><<<(NE / 16 * 32) / B, B, 0, stream>>>(ei, xb, dinv, z, xn);
        xin = xn;   // s==1 safely updates xA in place: last read of xin is in k_spmm
    }
}